// Softmax_70033736729139
// MI455X (gfx1250) — compile-verified
//
#include <hip/hip_runtime.h>
#include <hip/hip_bf16.h>

#define B    16
#define OUT  10000
#define IN   512
#define K    256

typedef __attribute__((ext_vector_type(2))) float v2f;
typedef __attribute__((ext_vector_type(8))) float v8f;

// ---------------------------------------------------------------------------
// Kernel 1: out[b,o] = bias[o] (default for all non-selected classes)
// ---------------------------------------------------------------------------
__global__ void init_out_kernel(const float* __restrict__ bias,
                                float* __restrict__ out) {
    int j = blockIdx.x * blockDim.x + threadIdx.x;
    if (j < B * OUT) out[j] = bias[j % OUT];
}

// ---------------------------------------------------------------------------
// Kernel 2: nw[o] = ||weight[o]||^2   (one wave32 per row)
// ---------------------------------------------------------------------------
__global__ void rownorms_kernel(const float* __restrict__ weight,
                                float* __restrict__ nw) {
    int wid  = threadIdx.x >> 5;
    int lane = threadIdx.x & 31;
    int r = blockIdx.x * 8 + wid;
    if (r >= OUT) return;
    const float* row = weight + (size_t)r * IN;
    float acc = 0.f;
    for (int i = lane; i < IN; i += 32) {
        float w = row[i];
        acc += w * w;
    }
    for (int off = 16; off > 0; off >>= 1)
        acc += __shfl_down(acc, off, 32);
    if (lane == 0) nw[r] = acc;
}

// ---------------------------------------------------------------------------
// Kernel 3: dis[b,o] = nw[labels[b]] + nw[o] - 2 * (sw[b] . w[o])
// One wave per block; each block computes a 16(b) x 16(o) tile using
// V_WMMA_F32_16X16X4_F32 chained over K=512 in steps of 4 (128 WMMAs).
//
// A (16x4 f32) layout per ISA: lanes 0-15 -> M=lane, VGPR0=K0 VGPR1=K1;
//                              lanes 16-31 -> M=lane-16, VGPR0=K2 VGPR1=K3.
// B (4x16) mirrors with N in the lane dimension.
// C/D (16x16 f32): lane -> N = lane&15; VGPR v -> M = v + (lane>=16 ? 8 : 0).
// ---------------------------------------------------------------------------
__global__ void dis_wmma_kernel(const float* __restrict__ weight,
                                const int* __restrict__ labels,
                                const float* __restrict__ nw,
                                float* __restrict__ dis) {
    int lane = threadIdx.x;          // 0..31 (wave32)
    int o0   = blockIdx.x * 16;      // 625 blocks cover OUT=10000 exactly
    int half = lane >> 4;            // 0: K=0/1   1: K=2/3
    int l15  = lane & 15;

    int rowA = labels[l15];
    const float* aPtr = weight + (size_t)rowA * IN + half * 2;
    const float* bPtr = weight + (size_t)(o0 + l15) * IN + half * 2;

    v8f c = {};
    #pragma unroll 4
    for (int k = 0; k < IN; k += 4) {
        v2f a  = *(const v2f*)(aPtr + k);
        v2f bb = *(const v2f*)(bPtr + k);
        // 8 args: (neg_a, A, neg_b, B, c_mod, C, reuse_a, reuse_b)
        c = __builtin_amdgcn_wmma_f32_16x16x4_f32(
                false, a, false, bb, (short)0, c, false, false);
    }

    int n = l15;
    float nwo = nw[o0 + n];
    #pragma unroll
    for (int v = 0; v < 8; ++v) {
        int m = v + half * 8;
        float nsw = nw[labels[m]];
        dis[(size_t)m * OUT + o0 + n] = nsw + nwo - 2.0f * c[v];
    }
}

// ---------------------------------------------------------------------------
// Kernel 4: per-b top-K (largest) selection via 4-pass MSB radix select.
// Only the SET of indices matters downstream (sum/max over K and the
// scatter are order-invariant), so a threshold + compaction is exact.
// ---------------------------------------------------------------------------
__device__ __forceinline__ unsigned f2key(float f) {
    unsigned u = __float_as_uint(f);
    return (u & 0x80000000u) ? ~u : (u | 0x80000000u);  // monotone: bigger f -> bigger key
}

__global__ void topk_select_kernel(const float* __restrict__ dis,
                                   int* __restrict__ sel) {
    int b = blockIdx.x;
    const float* row = dis + (size_t)b * OUT;
    __shared__ unsigned hist[256];
    __shared__ unsigned sh_prefix;
    __shared__ int sh_R;
    __shared__ int sh_fixed;
    __shared__ int sh_cnt;
    __shared__ int sh_tie;

    int tid = threadIdx.x;   // blockDim.x == 256
    if (tid == 0) { sh_prefix = 0; sh_R = K; sh_fixed = 0; sh_cnt = 0; sh_tie = 0; }
    __syncthreads();

    for (int pass = 0; pass < 4; ++pass) {
        hist[tid] = 0;
        __syncthreads();
        int fixed = sh_fixed;
        unsigned prefix = sh_prefix;
        int shift = 24 - 8 * pass;
        for (int o = tid; o < OUT; o += 256) {
            unsigned key = f2key(row[o]);
            bool match = (fixed == 0) || ((key >> (32 - fixed)) == prefix);
            if (match) atomicAdd(&hist[(key >> shift) & 0xFF], 1u);
        }
        __syncthreads();
        if (tid == 0) {
            int R = sh_R;
            unsigned cum = 0;
            int bsel = 0;
            for (int bin = 255; bin >= 0; --bin) {
                unsigned h = hist[bin];
                if (cum + h >= (unsigned)R) { bsel = bin; sh_R = R - (int)cum; break; }
                cum += h;
            }
            sh_prefix = (prefix << 8) | (unsigned)bsel;
            sh_fixed  = fixed + 8;
        }
        __syncthreads();
    }

    unsigned T = sh_prefix;      // exact 32-bit threshold key
    int tiesNeeded = sh_R;       // how many key==T elements to take
    __syncthreads();

    for (int o = tid; o < OUT; o += 256) {
        unsigned key = f2key(row[o]);
        if (key > T) {
            int p = atomicAdd(&sh_cnt, 1);
            sel[b * K + p] = o;
        } else if (key == T) {
            int t = atomicAdd(&sh_tie, 1);
            if (t < tiesNeeded) {
                int p = atomicAdd(&sh_cnt, 1);
                sel[b * K + p] = o;
            }
        }
    }
}

// ---------------------------------------------------------------------------
// Kernel 5: densities, mask threshold, masked dot products.
// Block per b, 512 threads (16 waves).
// Pass 1: thread i owns column i, loops over all K selected rows:
//   S[i] = sum_k dens, Mx[i] = max_k dens;  Tcol[i] = min(0.5*Mx, 0.1*max(S,1e-8))
//   (mask condition dens/S' >= min(0.5*Mx/S', 0.1)  <=>  dens >= Tcol[i])
// Pass 2: wave w handles rows k = w, w+16, ...; lanes stride columns,
//   recompute dens, apply mask, masked-dot with x, shfl-reduce, write out.
// ---------------------------------------------------------------------------
__global__ void mask_out_kernel(const float* __restrict__ x,
                                const float* __restrict__ mu,
                                const float* __restrict__ var,
                                const float* __restrict__ weight,
                                const float* __restrict__ bias,
                                const int* __restrict__ sel,
                                float* __restrict__ out) {
    int b   = blockIdx.x;
    int tid = threadIdx.x;            // 0..511
    __shared__ float Tcol[IN];
    __shared__ float xs[IN];
    __shared__ float mus[IN];
    __shared__ float vars[IN];
    __shared__ int   sl[K];

    if (tid < K) sl[tid] = sel[b * K + tid];

    int i = tid;                      // blockDim.x == IN == 512
    float m = mu[(size_t)b * IN + i];
    float v = var[(size_t)b * IN + i];
    mus[i]  = m;
    vars[i] = v;
    xs[i]   = x[(size_t)b * IN + i];
    float inv2v = 0.5f / v;
    __syncthreads();

    float S = 0.f, Mx = 0.f;
    for (int k = 0; k < K; ++k) {
        float w = weight[(size_t)sl[k] * IN + i];
        float t = w - m;
        float d = expf(-t * t * inv2v);
        S += d;
        Mx = fmaxf(Mx, d);
    }
    float Sc = fmaxf(S, 1e-8f);
    Tcol[i] = fminf(0.5f * Mx, 0.1f * Sc);
    __syncthreads();

    int wid  = tid >> 5;
    int lane = tid & 31;
    for (int kk = wid; kk < K; kk += 16) {
        int o = sl[kk];
        const float* wrow = weight + (size_t)o * IN;
        float acc = 0.f;
        for (int ii = lane; ii < IN; ii += 32) {
            float w = wrow[ii];
            float t = w - mus[ii];
            float d = expf(-t * t * (0.5f / vars[ii]));
            if (d >= Tcol[ii]) acc += xs[ii] * w;
        }
        for (int off = 16; off > 0; off >>= 1)
            acc += __shfl_down(acc, off, 32);
        if (lane == 0) out[(size_t)b * OUT + o] = acc + bias[o];
    }
}

// ---------------------------------------------------------------------------
// Launch: stream order provides all dependencies.
// Workspace layout (floats): nw[10240 pad] | dis[160000] | sel[4096 ints]
// ---------------------------------------------------------------------------
extern "C" void kernel_launch(void* const* d_in, const int* in_sizes, int n_in,
                              void* d_out, int out_size, void* d_ws, size_t ws_size,
                              hipStream_t stream) {
    (void)in_sizes; (void)n_in; (void)out_size; (void)ws_size;
    const float* x      = (const float*)d_in[0];
    const float* mu     = (const float*)d_in[1];
    const float* var    = (const float*)d_in[2];
    const int*   labels = (const int*)d_in[3];
    const float* weight = (const float*)d_in[4];
    const float* bias   = (const float*)d_in[5];
    float* out = (float*)d_out;

    float* wsf = (float*)d_ws;
    float* nw  = wsf;                         // 10000 (padded to 10240)
    float* dis = wsf + 10240;                 // B*OUT = 160000
    int*   sel = (int*)(wsf + 10240 + 160000);// B*K = 4096 ints

    init_out_kernel<<<(B * OUT + 255) / 256, 256, 0, stream>>>(bias, out);
    rownorms_kernel<<<(OUT + 7) / 8, 256, 0, stream>>>(weight, nw);
    dis_wmma_kernel<<<OUT / 16, 32, 0, stream>>>(weight, labels, nw, dis);
    topk_select_kernel<<<B, 256, 0, stream>>>(dis, sel);
    mask_out_kernel<<<B, IN, 0, stream>>>(x, mu, var, weight, bias, sel, out);
}